// Decoder_6511170421103
// MI455X (gfx1250) — compile-verified
//
#include <hip/hip_runtime.h>
#include <hip/hip_bf16.h>

#define N_BOX   1000
#define N_CLS   81
#define FLATN   (N_BOX * N_CLS)
#define TOPM    200
#define SORT_SZ 1024
#define NWORDS  32               // ceil(1000/32)
#define NMS_THR_F   0.5f
#define SCORE_THR_F 0.5f

// ---------- helpers ----------
__device__ __forceinline__ unsigned ordered_u32(float f) {
    unsigned u = __float_as_uint(f);
    return (u & 0x80000000u) ? ~u : (u | 0x80000000u);
}

// ---------- kernel 1: background flag (argmax over classes) ----------
__global__ void k_argmax(const float* __restrict__ probs, int* __restrict__ bg) {
    int n = blockIdx.x * blockDim.x + threadIdx.x;
    if (n >= N_BOX) return;
    const float* p = probs + (size_t)n * N_CLS;
    float best = p[0];
    int bi = 0;
    for (int c = 1; c < N_CLS; ++c) {
        float v = p[c];
        if (v > best) { best = v; bi = c; }   // first-max semantics (strict >)
    }
    bg[n] = (bi == 0) ? 1 : 0;
}

// ---------- kernel 2: delta decode + score, transposed to class-major ----------
__global__ void k_decode(const float* __restrict__ roi,
                         const float* __restrict__ deltas,
                         const float* __restrict__ probs,
                         const int* __restrict__ bg,
                         float4* __restrict__ dec,      // [C][N]
                         float* __restrict__ st) {      // [C][N]
    int e = blockIdx.x * blockDim.x + threadIdx.x;
    if (e >= FLATN) return;
    int n = e / N_CLS;
    int c = e % N_CLS;
    float a0 = roi[n * 4 + 0], a1 = roi[n * 4 + 1];
    float a2 = roi[n * 4 + 2], a3 = roi[n * 4 + 3];
    float aw = a3 - a1, ah = a2 - a0;
    float acx = a1 + 0.5f * aw, acy = a0 + 0.5f * ah;
    const float* d = deltas + (size_t)e * 4;
    float dy = d[0] * 0.1f, dx = d[1] * 0.1f;
    float dh = d[2] * 0.2f, dw = d[3] * 0.2f;
    float w  = expf(dw) * aw, h = expf(dh) * ah;
    float cx = dx * aw + acx, cy = dy * ah + acy;
    float y1 = cy - 0.5f * h, x1 = cx - 0.5f * w;
    dec[(size_t)c * N_BOX + n] = make_float4(y1, x1, y1 + h, x1 + w);
    st[(size_t)c * N_BOX + n] = bg[n] ? 0.0f : probs[e];
}

// ---------- kernel 3: per-class stable descending sort (bitonic on u64 keys) ----------
__global__ void k_sort(const float* __restrict__ st,
                       const float4* __restrict__ dec,
                       float* __restrict__ ssorted,
                       float4* __restrict__ bsorted) {
    __shared__ unsigned long long key[SORT_SZ];
    int c = blockIdx.x;
    int t = threadIdx.x;
    const float* sc = st + (size_t)c * N_BOX;
    unsigned long long k;
    if (t < N_BOX) {
        unsigned ord = ordered_u32(sc[t]);
        // ascending sort of this key == descending score, ties -> lower index first
        k = ((unsigned long long)(~ord) << 32) | (unsigned)t;
    } else {
        k = ~0ull;  // padding sinks to the end
    }
    key[t] = k;
    __syncthreads();
    for (unsigned kk = 2; kk <= SORT_SZ; kk <<= 1) {
        for (unsigned j = kk >> 1; j > 0; j >>= 1) {
            unsigned ixj = (unsigned)t ^ j;
            if (ixj > (unsigned)t) {
                unsigned long long a = key[t], b = key[ixj];
                bool asc = ((t & kk) == 0);
                if ((a > b) == asc) { key[t] = b; key[ixj] = a; }
            }
            __syncthreads();
        }
    }
    if (t < N_BOX) {
        unsigned idx = (unsigned)(key[t] & 0xFFFFFFFFu);
        ssorted[(size_t)c * N_BOX + t] = sc[idx];
        bsorted[(size_t)c * N_BOX + t] = dec[(size_t)c * N_BOX + idx];
    }
}

// ---------- kernel 4a: per-class pairwise suppression bitmask ----------
// mask[c][i][w] bit b set <=> j = w*32+b, j > i, IoU(i,j) > 0.5.
// Variant-independent, fully parallel; boxes staged via gfx1250 async DMA.
__global__ void k_mask(const float4* __restrict__ bsorted,
                       unsigned* __restrict__ mask) {
    __shared__ float4 sbox[N_BOX];
    __shared__ float  sarea[N_BOX];
    int c = blockIdx.x;
    int t = threadIdx.x;
    const float4* bptr = bsorted + (size_t)c * N_BOX;

    // Async DMA: 1000 x 16B global -> LDS (ASYNCcnt path)
    if (t < N_BOX) {
        unsigned lds_addr = (unsigned)(size_t)(&sbox[t]);  // low 32 bits == LDS offset
        unsigned goff     = (unsigned)(t * 16);
        asm volatile("global_load_async_to_lds_b128 %0, %1, %2"
                     :: "v"(lds_addr), "v"(goff), "s"(bptr)
                     : "memory");
    }
    asm volatile("s_wait_asynccnt 0x0" ::: "memory");
    __syncthreads();
    if (t < N_BOX) {
        float4 b = sbox[t];
        sarea[t] = (b.z - b.x) * (b.w - b.y);
    }
    __syncthreads();

    unsigned* mc = mask + (size_t)c * N_BOX * NWORDS;
    for (int idx = t; idx < N_BOX * NWORDS; idx += 1024) {
        int i = idx >> 5;
        int w = idx & 31;
        int jbase = w << 5;
        unsigned word = 0u;
        if (jbase + 31 > i) {                       // else all j <= i
            float4 bi = sbox[i];
            float  ai = sarea[i];
            int jstart = (jbase > i + 1) ? jbase : (i + 1);
            int jend   = (jbase + 32 < N_BOX) ? (jbase + 32) : N_BOX;
            for (int j = jstart; j < jend; ++j) {
                float4 bj = sbox[j];
                float ih = fminf(bi.z, bj.z) - fmaxf(bi.x, bj.x);
                float iw = fminf(bi.w, bj.w) - fmaxf(bi.y, bj.y);
                ih = fmaxf(ih, 0.0f);
                iw = fmaxf(iw, 0.0f);
                float inter = ih * iw;
                float uni   = ai + sarea[j] - inter;
                if (inter / (uni + 1e-9f) > NMS_THR_F) word |= 1u << (j - jbase);
            }
        }
        mc[idx] = word;
    }
}

// ---------- kernel 4b: greedy scan (single wave per variant, barrier-free loop) ----------
__global__ void k_scan(const float* __restrict__ ssorted,
                       const unsigned* __restrict__ mask,
                       float* __restrict__ ms) {          // [2][C][N]
    __shared__ float    ssc[N_BOX];
    __shared__ unsigned keep_lds[2][NWORDS];
    int c = blockIdx.x;
    int t = threadIdx.x;            // blockDim.x == 64: wave 0 -> variant 0, wave 1 -> variant 1
    const float* scptr = ssorted + (size_t)c * N_BOX;

    // Async DMA: 1000 floats (250 x 16B) global -> LDS
    for (int q = t; q < N_BOX / 4; q += 64) {
        unsigned lds_addr = (unsigned)(size_t)(&ssc[q * 4]);
        unsigned goff     = (unsigned)(q * 16);
        asm volatile("global_load_async_to_lds_b128 %0, %1, %2"
                     :: "v"(lds_addr), "v"(goff), "s"(scptr)
                     : "memory");
    }
    asm volatile("s_wait_asynccnt 0x0" ::: "memory");
    __syncthreads();

    int v = t >> 5;                 // variant (wave id)
    int w = t & 31;                 // keep-word owned by this lane
    {
        // init keep bits
        unsigned keepw = 0u;
        int base = w << 5;
        for (int b = 0; b < 32; ++b) {
            int j = base + b;
            if (j < N_BOX) {
                bool kb = (v == 0) ? (ssc[j] > SCORE_THR_F) : true;  // thr 0.5 vs -1e30
                if (kb) keepw |= 1u << b;
            }
        }
        const unsigned* mc = mask + (size_t)c * N_BOX * NWORDS;
        // sequential greedy scan: wave-lockstep, no barriers
        for (int i = 0; i < N_BOX; ++i) {
            unsigned ow = __shfl(keepw, i >> 5, 32);
            if ((ow >> (i & 31)) & 1u) {
                keepw &= ~mc[i * NWORDS + w];   // coalesced 128B row load
            }
        }
        // inclusive cumsum <= TOPM cap
        int pc = __popc(keepw);
        int pref = pc;
        for (int d = 1; d < 32; d <<= 1) {
            int y = __shfl_up(pref, d, 32);
            if (w >= d) pref += y;
        }
        int excl = pref - pc;       // kept boxes before this word
        if (excl >= TOPM) {
            keepw = 0u;
        } else if (excl + pc > TOPM) {
            int running = excl;
            unsigned nw = 0u;
            for (int b = 0; b < 32; ++b) {
                if ((keepw >> b) & 1u) {
                    ++running;
                    if (running <= TOPM) nw |= 1u << b;
                }
            }
            keepw = nw;
        }
        keep_lds[v][w] = keepw;
    }
    __syncthreads();
    // write masked scores for both variants
    for (int idx = t; idx < 2 * N_BOX; idx += 64) {
        int v2 = idx / N_BOX;
        int j  = idx % N_BOX;
        unsigned bit = (keep_lds[v2][j >> 5] >> (j & 31)) & 1u;
        ms[(size_t)v2 * FLATN + (size_t)c * N_BOX + j] = bit ? ssc[j] : -1.0f;
    }
}

// ---------- kernel 5: top-M selection per variant (JAX tie semantics) ----------
__global__ void k_topk(float* __restrict__ ms,            // [2][C][N], destroyed
                       const float4* __restrict__ bsorted,
                       float4* __restrict__ tb,            // [2][M]
                       float* __restrict__ tsv,            // [2][M]
                       float* __restrict__ tc,             // [2][M]
                       int* __restrict__ tvd) {            // [2]
    __shared__ unsigned long long warr[32];
    __shared__ unsigned long long rk0;
    __shared__ int scount;
    int v = blockIdx.x;
    int t = threadIdx.x;            // 1024 threads = 32 waves
    float* msv = ms + (size_t)v * FLATN;
    if (t == 0) scount = 0;

    // per-thread cached best over strided strip
    unsigned long long best = 0ull;
    for (int e = t; e < FLATN; e += 1024) {
        unsigned long long kk =
            ((unsigned long long)ordered_u32(msv[e]) << 32) | (unsigned)(~(unsigned)e);
        if (kk > best) best = kk;
    }

    for (int k = 0; k < TOPM; ++k) {
        // wave32 max-reduce (no barrier), then 32-entry hop through LDS
        unsigned long long wmax = best;
        for (int d = 16; d > 0; d >>= 1) {
            unsigned long long o = __shfl_xor(wmax, d, 32);
            if (o > wmax) wmax = o;
        }
        if ((t & 31) == 0) warr[t >> 5] = wmax;
        __syncthreads();
        if (t < 32) {
            unsigned long long x = warr[t];
            for (int d = 16; d > 0; d >>= 1) {
                unsigned long long o = __shfl_xor(x, d, 32);
                if (o > x) x = o;
            }
            if (t == 0) rk0 = x;
        }
        __syncthreads();
        unsigned long long win = rk0;
        int e = (int)(~(unsigned)(win & 0xFFFFFFFFu));
        if (t == (e & 1023)) {       // owner lane: emit, clobber, rescan own strip
            float s = msv[e];
            bool valid = s > 0.0f;
            int cls = e / N_BOX;
            tsv[v * TOPM + k] = valid ? s : 0.0f;
            tc [v * TOPM + k] = valid ? (float)cls : 0.0f;
            float4 bb = bsorted[e];
            tb [v * TOPM + k] = valid ? bb : make_float4(0.f, 0.f, 0.f, 0.f);
            if (valid) ++scount;
            msv[e] = -3.0e38f;
            best = 0ull;
            for (int e2 = t; e2 < FLATN; e2 += 1024) {
                unsigned long long kk =
                    ((unsigned long long)ordered_u32(msv[e2]) << 32) | (unsigned)(~(unsigned)e2);
                if (kk > best) best = kk;
            }
        }
        __syncthreads();
    }
    if (t == 0) tvd[v] = scount;
}

// ---------- kernel 6: variant select + pack output ----------
__global__ void k_final(const float4* __restrict__ tb,
                        const float* __restrict__ tsv,
                        const float* __restrict__ tc,
                        const int* __restrict__ tvd,
                        float* __restrict__ out) {
    __shared__ int sel;
    if (threadIdx.x == 0) sel = (tvd[0] == 0) ? 1 : 0;
    __syncthreads();
    int s = sel;
    for (int k = threadIdx.x; k < TOPM; k += blockDim.x) {
        float4 b = tb[s * TOPM + k];
        out[k * 4 + 0] = b.x;
        out[k * 4 + 1] = b.y;
        out[k * 4 + 2] = b.z;
        out[k * 4 + 3] = b.w;
        out[4 * TOPM + k] = tc [s * TOPM + k];   // labels
        out[5 * TOPM + k] = tsv[s * TOPM + k];   // scores
    }
}

extern "C" void kernel_launch(void* const* d_in, const int* in_sizes, int n_in,
                              void* d_out, int out_size, void* d_ws, size_t ws_size,
                              hipStream_t stream) {
    const float* roi    = (const float*)d_in[0];   // (1,1000,4)
    const float* deltas = (const float*)d_in[1];   // (1,1000,81,4)
    const float* probs  = (const float*)d_in[2];   // (1,1000,81)
    float* out = (float*)d_out;                    // 800 boxes + 200 labels + 200 scores

    char* ws = (char*)d_ws;
    size_t cur = 0;
    auto alloc = [&](size_t bytes) -> char* {
        cur = (cur + 255) & ~(size_t)255;
        char* p = ws + cur;
        cur += bytes;
        return p;
    };
    int*      bg      = (int*)     alloc(N_BOX * sizeof(int));
    float4*   dec     = (float4*)  alloc((size_t)FLATN * sizeof(float4));
    float*    st      = (float*)   alloc((size_t)FLATN * sizeof(float));
    float*    ssorted = (float*)   alloc((size_t)FLATN * sizeof(float));
    float4*   bsorted = (float4*)  alloc((size_t)FLATN * sizeof(float4));
    unsigned* maskbuf = (unsigned*)alloc((size_t)N_CLS * N_BOX * NWORDS * sizeof(unsigned));
    float*    msbuf   = (float*)   alloc((size_t)2 * FLATN * sizeof(float));
    float4*   tb      = (float4*)  alloc((size_t)2 * TOPM * sizeof(float4));
    float*    tsv     = (float*)   alloc((size_t)2 * TOPM * sizeof(float));
    float*    tc      = (float*)   alloc((size_t)2 * TOPM * sizeof(float));
    int*      tvd     = (int*)     alloc(2 * sizeof(int));

    k_argmax<<<(N_BOX + 255) / 256, 256, 0, stream>>>(probs, bg);
    k_decode<<<(FLATN + 255) / 256, 256, 0, stream>>>(roi, deltas, probs, bg, dec, st);
    k_sort  <<<N_CLS, SORT_SZ, 0, stream>>>(st, dec, ssorted, bsorted);
    k_mask  <<<N_CLS, 1024, 0, stream>>>(bsorted, maskbuf);
    k_scan  <<<N_CLS, 64, 0, stream>>>(ssorted, maskbuf, msbuf);
    k_topk  <<<2, 1024, 0, stream>>>(msbuf, bsorted, tb, tsv, tc, tvd);
    k_final <<<1, 256, 0, stream>>>(tb, tsv, tc, tvd, out);
}